// AdjointConformTNN_38749194944738
// MI455X (gfx1250) — compile-verified
//
#include <hip/hip_runtime.h>
#include <hip/hip_bf16.h>

typedef float v2f __attribute__((ext_vector_type(2)));
typedef float v8f __attribute__((ext_vector_type(8)));

#define NFEAT 7
#define NVIRT 4
#define NCOND 15
#define NIN   (NFEAT + NVIRT)   // 11

// Deterministic index tables from _build_index_tables() (triu of 6x6 adjacency):
// temps_indexer[i] = all j != i in 0..5 ; adj_idx[i][j] = adj[i][temps_indexer[i][j]]
__device__ __constant__ int TIX[4][5] = {
    {1, 2, 3, 4, 5}, {0, 2, 3, 4, 5}, {0, 1, 3, 4, 5}, {0, 1, 2, 4, 5}};
__device__ __constant__ int AJX[4][5] = {
    {0, 1, 2, 3, 4}, {0, 5, 6, 7, 8}, {1, 5, 9, 10, 11}, {2, 6, 9, 12, 13}};

// Address-select feature fetch: k is clamped to NIN-1 (the fused B-matrix has
// zero weights for k >= NIN, so the padded A element may be any finite value).
// Selecting the *address* keeps this a single unpredicated load -> EXEC stays
// all-ones through the WMMA region (ISA requirement, and avoids saveexec).
__device__ __forceinline__ const float* feat_addr(const float* __restrict__ u_step,
                                                  const float* __restrict__ x,
                                                  int row, int k) {
  k = min(k, NIN - 1);
  return (k < NFEAT) ? (u_step + row * NFEAT + k)
                     : (x + row * NVIRT + (k - NFEAT));
}

__device__ __forceinline__ float fast_tanh(float z) {
  // tanh(z) = 1 - 2/(exp(2z)+1); saturates correctly at +-1
  return 1.0f - 2.0f / (__expf(2.0f * z) + 1.0f);
}

__global__ __launch_bounds__(256) void tnn_wmma_kernel(
    const float* __restrict__ t_ptr, const float* __restrict__ x,
    const float* __restrict__ u, const float* __restrict__ caps,
    const float* __restrict__ cw1, const float* __restrict__ cb1,
    const float* __restrict__ cw2, const float* __restrict__ cb2,
    const float* __restrict__ pw1, const float* __restrict__ pb1,
    const float* __restrict__ pw2, const float* __restrict__ pb2,
    const int* __restrict__ temp_idcs, float* __restrict__ out, int B) {
  // per-wave private LDS slab: 2 halves x 16 rows x 17 (padded) floats
  __shared__ float lds[8][2 * 16 * 17];

  const int lane = threadIdx.x & 31;
  const int wib  = threadIdx.x >> 5;
  const int waveId = blockIdx.x * (blockDim.x >> 5) + wib;
  const int nWaves = gridDim.x * (blockDim.x >> 5);

  int step = (int)(t_ptr[0] * 2.0f);
  step = max(0, min(3, step));
  const float* u_step = u + (size_t)step * (size_t)B * NFEAT;

  // ---- per-lane B-matrix (fused [cw1 | pw1 | 0] : 11(K) x 16(N), K padded to 12) ----
  // B layout: lane holds column n = lane&15; K-pair {0,1} (lanes 0-15) or {2,3} (16-31)
  const int n  = lane & 15;
  const int kb = (lane < 16) ? 0 : 2;
  v2f bm[3];
#pragma unroll
  for (int kt = 0; kt < 3; ++kt) {
    float w0 = 0.0f, w1 = 0.0f;
    int k0 = 4 * kt + kb, k1 = k0 + 1;
    if (k0 < NIN) w0 = (n < 2) ? cw1[k0 * 2 + n] : ((n < 6) ? pw1[k0 * 4 + (n - 2)] : 0.0f);
    if (k1 < NIN) w1 = (n < 2) ? cw1[k1 * 2 + n] : ((n < 6) ? pw1[k1 * 4 + (n - 2)] : 0.0f);
    bm[kt].x = w0;
    bm[kt].y = w1;
  }

  // uniform small tables (scalar-load friendly: uniform addresses)
  const float b1c0 = cb1[0], b1c1 = cb1[1];
  float b1p[4], ec[4], pb2v[4];
#pragma unroll
  for (int j = 0; j < 4; ++j) {
    b1p[j]  = pb1[j];
    ec[j]   = __expf(caps[j]);
    pb2v[j] = pb2[j];
  }
  const int tid0 = temp_idcs[0], tid1 = temp_idcs[1];

  const int nTiles = (B + 31) >> 5;
  for (int tile = waveId; tile < nTiles; tile += nWaves) {
    const int base = tile << 5;

    // -------- phase 1: fused layer-1 GEMM via V_WMMA_F32_16X16X4_F32 --------
#pragma unroll
    for (int half = 0; half < 2; ++half) {
      const int m = lane & 15;
      int row = base + half * 16 + m;
      row = min(row, B - 1);
      v8f acc = {0.f, 0.f, 0.f, 0.f, 0.f, 0.f, 0.f, 0.f};
#pragma unroll
      for (int kt = 0; kt < 3; ++kt) {
        v2f a;
        a.x = *feat_addr(u_step, x, row, 4 * kt + kb);
        a.y = *feat_addr(u_step, x, row, 4 * kt + kb + 1);
        acc = __builtin_amdgcn_wmma_f32_16x16x4_f32(
            /*neg_a=*/false, a, /*neg_b=*/false, bm[kt],
            /*c_mod=*/(short)0, acc, /*reuse_a=*/false, /*reuse_b=*/false);
      }
      // D layout: lane holds col n=lane&15, rows v + 8*(lane>=16), v=0..7
      const int mBase = (lane >> 4) << 3;
#pragma unroll
      for (int v = 0; v < 8; ++v)
        lds[wib][(half * 16 + mBase + v) * 17 + n] = acc[v];
    }

    // -------- phase 2: one row per lane, scalar finish --------
    const int rr = base + lane;
    const int rc = min(rr, B - 1);

    float h[6];
#pragma unroll
    for (int j = 0; j < 6; ++j) h[j] = lds[wib][lane * 17 + j];

    const float4 xv = *(const float4*)(x + (size_t)rc * NVIRT);

    const float hc0 = fast_tanh(h[0] + b1c0);
    const float hc1 = fast_tanh(h[1] + b1c1);
    float hp[4];
#pragma unroll
    for (int j = 0; j < 4; ++j) hp[j] = fast_tanh(h[2 + j] + b1p[j]);

    // conducts = elu(hc @ cw2 + cb2) + 1   (z>0 ? z+1 : exp(z))
    float cond[NCOND];
#pragma unroll
    for (int c = 0; c < NCOND; ++c) {
      float z = fmaf(hc0, cw2[c], fmaf(hc1, cw2[NCOND + c], cb2[c]));
      cond[c] = (z > 0.0f) ? (z + 1.0f) : __expf(z);
    }

    // power_loss = sigmoid(hp @ pw2 + pb2)
    float pl[4];
#pragma unroll
    for (int i = 0; i < 4; ++i) {
      float z = pb2v[i];
#pragma unroll
      for (int j = 0; j < 4; ++j) z = fmaf(hp[j], pw2[j * 4 + i], z);
      pl[i] = 1.0f / (1.0f + __expf(-z));
    }

    float temps[6];
    temps[0] = xv.x; temps[1] = xv.y; temps[2] = xv.z; temps[3] = xv.w;
    temps[4] = u_step[rc * NFEAT + tid0];
    temps[5] = u_step[rc * NFEAT + tid1];

    float4 o;
    float* op = (float*)&o;
#pragma unroll
    for (int i = 0; i < 4; ++i) {
      float td = 0.0f;
#pragma unroll
      for (int j = 0; j < 5; ++j)
        td = fmaf(temps[TIX[i][j]] - temps[i], cond[AJX[i][j]], td);
      float val = ec[i] * (td + pl[i]);
      op[i] = fminf(1.0f, fmaxf(-1.0f, val));
    }

    if (rr < B) *(float4*)(out + (size_t)rr * NVIRT) = o;
  }
}

extern "C" void kernel_launch(void* const* d_in, const int* in_sizes, int n_in,
                              void* d_out, int out_size, void* d_ws, size_t ws_size,
                              hipStream_t stream) {
  const float* t_ptr = (const float*)d_in[0];
  const float* x     = (const float*)d_in[1];
  const float* u     = (const float*)d_in[2];
  const float* caps  = (const float*)d_in[3];
  const float* cw1   = (const float*)d_in[4];
  const float* cb1   = (const float*)d_in[5];
  const float* cw2   = (const float*)d_in[6];
  const float* cb2   = (const float*)d_in[7];
  const float* pw1   = (const float*)d_in[8];
  const float* pb1   = (const float*)d_in[9];
  const float* pw2   = (const float*)d_in[10];
  const float* pb2   = (const float*)d_in[11];
  // d_in[12] = temps_indexer, d_in[13] = adj_idx  (deterministic; hardcoded in-kernel)
  const int* temp_idcs = (const int*)d_in[14];

  const int B = in_sizes[1] / NVIRT;

  const int threads = 256;
  int tiles = (B + 31) / 32;
  int blocks = (tiles + 7) / 8;                // 8 waves per block, 1 tile/wave min
  if (blocks > 1024) blocks = 1024;            // grid-stride beyond this
  if (blocks < 1) blocks = 1;

  tnn_wmma_kernel<<<blocks, threads, 0, stream>>>(
      t_ptr, x, u, caps, cw1, cb1, cw2, cb2, pw1, pb1, pw2, pb2,
      temp_idcs, (float*)d_out, B);
}